// STPMLP_36472862277840
// MI455X (gfx1250) — compile-verified
//
#include <hip/hip_runtime.h>
#include <hip/hip_bf16.h>
#include <math.h>

typedef __attribute__((ext_vector_type(2))) float v2f;
typedef __attribute__((ext_vector_type(8))) float v8f;

#define D_MODEL 512
#define D_FF    2048
#define BATCH   4
#define SEQ     1024
#define M_TOTAL (BATCH * SEQ) /* 4096 */

// ---------------------------------------------------------------------------
// GEMM: C[M][N] = A[M][K] * B[N][K]^T + bias[N]   (all f32, exact)
// Uses V_WMMA_F32_16X16X4_F32. Block tile 64(M) x 32(N), 8 waves, each wave
// one 16x16 WMMA tile. K staged through LDS in chunks of 32.
//
// f32 WMMA operand layouts (ISA 7.12.2):
//   A 16x4 : lane l -> M = l%16 ; VGPR v holds K = 2*(l/16) + v
//   B 4x16 : lane l -> N = l%16 ; VGPR v holds K = 2*(l/16) + v   (mirror of A)
//   C/D    : lane l -> N = l%16 ; VGPR i holds M = i + 8*(l/16)
// ---------------------------------------------------------------------------
template <int K>
__global__ __launch_bounds__(256) void gemm_wmma_f32(
    const float* __restrict__ A, const float* __restrict__ Bm,
    const float* __restrict__ bias, float* __restrict__ C, int nBlocksN) {
  __shared__ float Asl[64][32];  // [m][k]
  __shared__ float Bsl[32][32];  // [n][k]

  const int bm = blockIdx.x / nBlocksN;
  const int bn = blockIdx.x % nBlocksN;
  const int M0 = bm * 64, N0 = bn * 32;
  const int Ntot = nBlocksN * 32;

  const int tid = threadIdx.x;
  const int wave = tid >> 5, lane = tid & 31;
  const int wm = wave & 3, wn = wave >> 2;  // 4 x 2 wave grid
  const int nl = lane & 15, half = lane >> 4;

  v8f acc = {};

  for (int k0 = 0; k0 < K; k0 += 32) {
    // Cooperative stage: A chunk 64x32 (512 float4), B chunk 32x32 (256 float4)
    {
      int idx = tid;
#pragma unroll
      for (int r = 0; r < 2; ++r) {
        const int row = idx >> 3, c4 = (idx & 7) << 2;
        *(float4*)&Asl[row][c4] =
            *(const float4*)&A[(size_t)(M0 + row) * K + k0 + c4];
        idx += 256;
      }
      const int row = tid >> 3, c4 = (tid & 7) << 2;
      *(float4*)&Bsl[row][c4] =
          *(const float4*)&Bm[(size_t)(N0 + row) * K + k0 + c4];
    }
    __syncthreads();

#pragma unroll
    for (int kk = 0; kk < 32; kk += 4) {
      const int ka = kk + (half << 1);
      v2f a, b;
      a.x = Asl[wm * 16 + nl][ka];
      a.y = Asl[wm * 16 + nl][ka + 1];
      b.x = Bsl[wn * 16 + nl][ka];
      b.y = Bsl[wn * 16 + nl][ka + 1];
      acc = __builtin_amdgcn_wmma_f32_16x16x4_f32(false, a, false, b, (short)0,
                                                  acc, false, false);
    }
    __syncthreads();
  }

  const int ng = N0 + wn * 16 + nl;
  const float bv = bias[ng];
#pragma unroll
  for (int i = 0; i < 8; ++i) {
    const int mg = M0 + wm * 16 + i + 8 * half;
    C[(size_t)mg * Ntot + ng] = acc[i] + bv;
  }
}

// ---------------------------------------------------------------------------
// Sequential Hebbian scan. One wave owns one (b, f) row of the fast-weight
// state F[b,f,0:512]: 16 f32 per lane, resident in VGPRs for all 1024 steps
// (zero state traffic). Per step: shared x_t from LDS (double buffered, one
// barrier/step), 16-FMA partial dot, wave butterfly reduction, exact GELU,
// store h for the deferred output GEMM, rank-1 VGPR state update.
//   block = 8 waves = 8 consecutive f rows of one batch; grid = 4*256 blocks.
// ---------------------------------------------------------------------------
__global__ __launch_bounds__(256) void stp_scan(
    const float* __restrict__ x,         // [B][L][D]
    const float* __restrict__ static_h,  // [B*L][D_FF]
    const float* __restrict__ Lambda,    // [D_FF][D]
    const float* __restrict__ Gamma,     // [D_FF][D]
    float* __restrict__ H)               // [B*L][D_FF] = gelu(h_pre)
{
  __shared__ float xs[2][D_MODEL];

  const int tid = threadIdx.x;
  const int lane = tid & 31;
  const int wave = tid >> 5;
  const int b = blockIdx.x >> 8;                       // 256 blocks per batch
  const int f = ((blockIdx.x & 255) << 3) + wave;      // this wave's d_ff row

  float F[16], ret[16], gam[16];
#pragma unroll
  for (int k = 0; k < 16; ++k) {
    const int d = (k << 5) + lane;                     // strided: conflict-free
    F[k] = 0.0f;
    // retention = 1 - sigmoid(z) = 1 / (1 + e^z)
    ret[k] = 1.0f / (1.0f + expf(Lambda[(size_t)f * D_MODEL + d]));
    gam[k] = Gamma[(size_t)f * D_MODEL + d];
  }

  for (int t = 0; t < SEQ; ++t) {
    const int buf = t & 1;
    const size_t m = (size_t)(b * SEQ + t);
    // stage x_t (512 floats) cooperatively
    xs[buf][tid] = x[m * D_MODEL + tid];
    xs[buf][tid + 256] = x[m * D_MODEL + tid + 256];
    __syncthreads();

    float xv[16];
    float acc = 0.0f;
#pragma unroll
    for (int k = 0; k < 16; ++k) {
      xv[k] = xs[buf][(k << 5) + lane];
      acc = fmaf(F[k], xv[k], acc);
    }
    // wave32 butterfly: all lanes end with the full 512-wide dot product
#pragma unroll
    for (int off = 16; off > 0; off >>= 1) acc += __shfl_xor(acc, off, 32);

    const float hp = acc + static_h[m * D_FF + f];     // wave-uniform load
    const float h = 0.5f * hp * (1.0f + erff(hp * 0.70710678118654752f));
    if (lane == 0) H[m * D_FF + f] = h;

    // F = retention*F + gamma * (h_pre * x^T)
#pragma unroll
    for (int k = 0; k < 16; ++k)
      F[k] = fmaf(ret[k], F[k], (gam[k] * hp) * xv[k]);
  }
}

// ---------------------------------------------------------------------------
extern "C" void kernel_launch(void* const* d_in, const int* in_sizes, int n_in,
                              void* d_out, int out_size, void* d_ws,
                              size_t ws_size, hipStream_t stream) {
  (void)in_sizes; (void)n_in; (void)out_size; (void)ws_size;
  const float* x  = (const float*)d_in[0];   // (B, L, D)
  const float* W1 = (const float*)d_in[1];   // (D_FF, D)
  const float* b1 = (const float*)d_in[2];   // (D_FF,)
  const float* W2 = (const float*)d_in[3];   // (D, D_FF)
  const float* b2 = (const float*)d_in[4];   // (D,)
  const float* Lr = (const float*)d_in[5];   // (D_FF, D)
  const float* Gm = (const float*)d_in[6];   // (D_FF, D)
  float* out = (float*)d_out;                // (B, L, D)

  float* static_h = (float*)d_ws;                         // 32 MB
  float* H = static_h + (size_t)M_TOTAL * D_FF;           // 32 MB

  // 1) static_h = x @ W1^T + b1   (M=4096, N=2048, K=512)
  gemm_wmma_f32<D_MODEL>
      <<<(M_TOTAL / 64) * (D_FF / 32), 256, 0, stream>>>(x, W1, b1, static_h,
                                                         D_FF / 32);
  // 2) sequential fast-weight scan; emits H = gelu(h_pre)
  stp_scan<<<BATCH * (D_FF / 8), 256, 0, stream>>>(x, static_h, Lr, Gm, H);
  // 3) out = H @ W2^T + b2        (M=4096, N=512, K=2048)
  gemm_wmma_f32<D_FF>
      <<<(M_TOTAL / 64) * (D_MODEL / 32), 256, 0, stream>>>(H, W2, b2, out,
                                                            D_MODEL / 32);
}